// diff_lpc2rc_79336635892366
// MI455X (gfx1250) — compile-verified
//
#include <hip/hip_runtime.h>

#define LPC_N 16

// Native clang vector type: required for __builtin_nontemporal_load/store,
// and guarantees b128 global load/store lowering.
typedef float v4f __attribute__((ext_vector_type(4)));

// Levinson step-down (LPC -> reflection coefficients).
// One thread per 16-float row: 4x b128 NT loads, fully-unrolled in-register
// recursion, 4x b128 NT stores. Memory-bound: ~262 MB streamed once
// (~11.3 us floor at 23.3 TB/s). v_rcp_f32 instead of IEEE divide keeps
// VALU work (~250 ops/row) under the memory roof.
__global__ __launch_bounds__(256) void diff_lpc2rc_kernel(
    const float* __restrict__ in, float* __restrict__ out, int nrows)
{
    int row = blockIdx.x * 256 + threadIdx.x;
    if (row >= nrows) return;

    const v4f* ip = reinterpret_cast<const v4f*>(in) + (size_t)row * 4;
    v4f*       op = reinterpret_cast<v4f*>(out)      + (size_t)row * 4;

    // Streaming loads: data is touched exactly once -> non-temporal hint
    // (262 MB total > 192 MB L2, so keep it out of the cache).
    v4f v0 = __builtin_nontemporal_load(ip + 0);
    v4f v1 = __builtin_nontemporal_load(ip + 1);
    v4f v2 = __builtin_nontemporal_load(ip + 2);
    v4f v3 = __builtin_nontemporal_load(ip + 3);

    float r[LPC_N] = { v0.x, v0.y, v0.z, v0.w,
                       v1.x, v1.y, v1.z, v1.w,
                       v2.x, v2.y, v2.z, v2.w,
                       v3.x, v3.y, v3.z, v3.w };

    // Step-down recursion. At step i: k = r[16-i], update r[0 .. 15-i].
    // One v_rcp_f32 per step (denominator ~1.0, ~1 ulp accurate) instead of
    // the 8-op IEEE divide sequence: saves ~105 VALU ops per row.
#pragma unroll
    for (int i = 1; i < LPC_N; ++i) {
        const int h  = LPC_N - i;      // head length
        const float ki  = r[h];
        const float inv = __builtin_amdgcn_rcpf(1.0f - ki * ki);
        // Symmetric pair update handles the reversal with no temp array.
#pragma unroll
        for (int m = 0; m < h / 2; ++m) {
            const float a = r[m];
            const float b = r[h - 1 - m];
            r[m]         = (a - ki * b) * inv;
            r[h - 1 - m] = (b - ki * a) * inv;
        }
        if (h & 1) {
            const int  mid = h / 2;
            const float a  = r[mid];
            r[mid] = (a - ki * a) * inv;
        }
    }

    v4f w0 = { r[0],  r[1],  r[2],  r[3]  };
    v4f w1 = { r[4],  r[5],  r[6],  r[7]  };
    v4f w2 = { r[8],  r[9],  r[10], r[11] };
    v4f w3 = { r[12], r[13], r[14], r[15] };

    __builtin_nontemporal_store(w0, op + 0);
    __builtin_nontemporal_store(w1, op + 1);
    __builtin_nontemporal_store(w2, op + 2);
    __builtin_nontemporal_store(w3, op + 3);
}

extern "C" void kernel_launch(void* const* d_in, const int* in_sizes, int n_in,
                              void* d_out, int out_size, void* d_ws, size_t ws_size,
                              hipStream_t stream)
{
    (void)n_in; (void)d_ws; (void)ws_size; (void)out_size;
    const float* in  = (const float*)d_in[0];
    float*       out = (float*)d_out;

    const int nrows  = in_sizes[0] / LPC_N;   // 128 * 16000 = 2,048,000
    const int block  = 256;                   // 8 wave32 waves per block
    const int grid   = (nrows + block - 1) / block;

    diff_lpc2rc_kernel<<<grid, block, 0, stream>>>(in, out, nrows);
}